// DigitCapsLayer_23416161697957
// MI455X (gfx1250) — compile-verified
//
#include <hip/hip_runtime.h>

// CDNA5 (gfx1250) fused CapsNet dynamic routing.
// u_hat never materialized (189MB avoided): recomputed per routing iteration with
// V_WMMA_F32_16X16X4_F32 in transposed form (M=o, N=batch) so each lane owns one
// batch's half o-vector -> softmax/routing are lane-local (+1 xor-16 shuffle).
// b_ij = u_hat . (sum of previous v's) -> only Vsum[b,j,o] kept in LDS.
// W is pre-packed once into the exact WMMA A-operand lane layout so the hot loop
// does one b128 load per (route, capsule) instead of 4 strided b32 loads.

typedef __attribute__((ext_vector_type(2))) float v2f;
typedef __attribute__((ext_vector_type(4))) float v4f;
typedef __attribute__((ext_vector_type(8))) float v8f;

#define R_TOT 1152
#define JC 10
#define OC 16
#define IC 8
#define BT 16      // batch tile = WMMA N
#define WAVES 8
#define ITERS 3
#define WP_ELEMS (R_TOT * JC * 32 * 4)   // packed W: 1,474,560 floats = 5.9 MB

// Pack W[r,j,i,o] -> Wp[r,j,lane,q] where lane=(h<<4)|lo and
//   q=0: i=2h   q=1: i=2h+1   q=2: i=4+2h   q=3: i=5+2h   (o = lo)
// so a lane's two K-step A operands are one contiguous float4.
__global__ void pack_W_kernel(const float* __restrict__ W, float* __restrict__ Wp) {
    const int idx = blockIdx.x * 256 + threadIdx.x;
    if (idx >= WP_ELEMS) return;
    const int q    = idx & 3;
    const int lane = (idx >> 2) & 31;
    const int rj   = idx >> 7;           // r*JC + j
    const int h = lane >> 4, lo = lane & 15;
    const int i = (q & 1) + 2 * h + 4 * (q >> 1);
    Wp[idx] = W[((size_t)rj * IC + i) * OC + lo];
}

template <bool PACKED>
__launch_bounds__(256, 1)
__global__ void caps_route_kernel(const float* __restrict__ x,
                                  const float* __restrict__ W,
                                  const float* __restrict__ Wp,
                                  float* __restrict__ out)
{
    __shared__ float s_lds[BT * JC * OC];     // s_j accumulator (zeroed per iter)
    __shared__ float vsum_lds[BT * JC * OC];  // sum of v over past iterations

    const int tid  = threadIdx.x;
    const int wave = tid >> 5;                // wave32
    const int lane = tid & 31;
    const int half = lane >> 4;               // 0: o = t, 1: o = t+8
    const int lo   = lane & 15;               // batch-in-tile (WMMA N index)
    const int b0   = blockIdx.x * BT;

    for (int i = tid; i < BT * JC * OC; i += 256) vsum_lds[i] = 0.0f;
    __syncthreads();

    for (int it = 0; it < ITERS; ++it) {
        for (int i = tid; i < BT * JC * OC; i += 256) s_lds[i] = 0.0f;
        __syncthreads();

        // Per-lane cache of Vsum: vs[j][t] = Vsum[b=lo][j][o=t+8*half]
        float vs[JC][8];
        #pragma unroll
        for (int j = 0; j < JC; ++j)
            #pragma unroll
            for (int t = 0; t < 8; ++t)
                vs[j][t] = vsum_lds[(lo * JC + j) * OC + (t + 8 * half)];

        float sacc[JC][8];
        #pragma unroll
        for (int j = 0; j < JC; ++j)
            #pragma unroll
            for (int t = 0; t < 8; ++t) sacc[j][t] = 0.0f;

        for (int r = wave; r < R_TOT; r += WAVES) {
            // B operand = x^T tile (K=i rows, N=batch cols), two K-steps of 4.
            // B layout: VGPR0 = K rows {ks+0 | ks+2} by lane half, VGPR1 = {ks+1 | ks+3}.
            const float* xr = x + ((size_t)(b0 + lo) * R_TOT + r) * IC;
            v2f bx0 = *(const v2f*)(xr + 0 + 2 * half);
            v2f bx1 = *(const v2f*)(xr + 4 + 2 * half);

            // u[j] = W[r,j]^T (16o x 8i) x x^T (8i x 16b); C layout: lane owns
            // (b = b0+lo, o = t+8*half) for t = 0..7 across the 8 C VGPRs.
            v8f u[JC];
            #pragma unroll
            for (int j = 0; j < JC; ++j) {
                v2f a0, a1;                       // A = W^T: M=o=lane%16, K=i
                if (PACKED) {
                    const v4f w4 = *(const v4f*)(Wp +
                        (((size_t)r * JC + j) * 32 + lane) * 4);
                    a0.x = w4.x; a0.y = w4.y;
                    a1.x = w4.z; a1.y = w4.w;
                } else {
                    const float* wj = W + ((size_t)r * JC + j) * IC * OC;
                    a0.x = wj[(0 + 2 * half) * OC + lo];
                    a0.y = wj[(1 + 2 * half) * OC + lo];
                    a1.x = wj[(4 + 2 * half) * OC + lo];
                    a1.y = wj[(5 + 2 * half) * OC + lo];
                }
                v8f c = {};
                c = __builtin_amdgcn_wmma_f32_16x16x4_f32(false, a0, false, bx0,
                                                          (short)0, c, false, false);
                c = __builtin_amdgcn_wmma_f32_16x16x4_f32(false, a1, false, bx1,
                                                          (short)0, c, false, false);
                u[j] = c;
            }

            // logits: b_ij = u_hat . Vsum ; per-lane 8 FMAs + one xor-16 shuffle
            float cj[JC];
            float mx = -3.402823466e38f;
            #pragma unroll
            for (int j = 0; j < JC; ++j) {
                float p = 0.0f;
                #pragma unroll
                for (int t = 0; t < 8; ++t) p = fmaf(u[j][t], vs[j][t], p);
                p += __shfl_xor(p, 16, 32);       // add other o-half
                cj[j] = p;
                mx = fmaxf(mx, p);
            }
            // softmax over j (iter 0: all zero -> uniform 0.1, matches reference)
            float ssum = 0.0f;
            #pragma unroll
            for (int j = 0; j < JC; ++j) { cj[j] = __expf(cj[j] - mx); ssum += cj[j]; }
            const float inv = 1.0f / ssum;
            #pragma unroll
            for (int j = 0; j < JC; ++j) {
                const float cc = cj[j] * inv;
                #pragma unroll
                for (int t = 0; t < 8; ++t)
                    sacc[j][t] = fmaf(cc, u[j][t], sacc[j][t]);
            }
        }

        // Cross-wave reduction of s into LDS (conflict-free within a wave)
        #pragma unroll
        for (int j = 0; j < JC; ++j)
            #pragma unroll
            for (int t = 0; t < 8; ++t)
                atomicAdd(&s_lds[(lo * JC + j) * OC + (t + 8 * half)], sacc[j][t]);
        __syncthreads();

        // squash + Vsum update (+ final output write) by 160 threads
        if (tid < BT * JC) {
            const int bl = tid / JC, j = tid % JC;
            float sv[OC];
            float sq = 0.0f;
            #pragma unroll
            for (int o = 0; o < OC; ++o) {
                sv[o] = s_lds[(bl * JC + j) * OC + o];
                sq = fmaf(sv[o], sv[o], sq);
            }
            const float scale = (sq / (1.0f + sq)) * rsqrtf(sq + 1e-8f);
            #pragma unroll
            for (int o = 0; o < OC; ++o) {
                const float v = scale * sv[o];
                vsum_lds[(bl * JC + j) * OC + o] += v;
                if (it == ITERS - 1)
                    out[((size_t)(b0 + bl) * JC + j) * OC + o] = v;
            }
        }
        __syncthreads();
    }
}

extern "C" void kernel_launch(void* const* d_in, const int* in_sizes, int n_in,
                              void* d_out, int out_size, void* d_ws, size_t ws_size,
                              hipStream_t stream) {
    (void)in_sizes; (void)n_in; (void)out_size;
    const float* x = (const float*)d_in[0];   // [256,1152,8]
    const float* W = (const float*)d_in[1];   // [1152,10,8,16]
    float* out = (float*)d_out;               // [256,10,16]

    const size_t need = (size_t)WP_ELEMS * sizeof(float);
    dim3 grid(256 / BT), block(256);
    if (ws_size >= need) {
        float* Wp = (float*)d_ws;
        pack_W_kernel<<<(WP_ELEMS + 255) / 256, 256, 0, stream>>>(W, Wp);
        caps_route_kernel<true><<<grid, block, 0, stream>>>(x, W, Wp, out);
    } else {
        caps_route_kernel<false><<<grid, block, 0, stream>>>(x, W, nullptr, out);
    }
}